// MazeCollision_65584150610356
// MI455X (gfx1250) — compile-verified
//
#include <hip/hip_runtime.h>

// ---------------------------------------------------------------------------
// MazeCollision on MI455X (gfx1250, wave32).
// d2 = |p|^2 + |c|^2 - 2 p.c computed per 16x16 tile with one
// V_WMMA_F32_16X16X4_F32:  A = pos rows (K=2 real, K=2 zero-pad),
// B = -2 * cell centers, C preloaded with |p|^2_row + |c|^2_col,
// so D is d2 directly.
//
// Exactness-preserving tile culling: a 16-col tile is a strip of ONE maze
// row; relu(rad - d) is identically zero unless some position is within rad
// of the strip. Conservative bbox test + ballot -> wave-uniform s_cbranch
// skips WMMA + load + epilogue for ~94% of tiles (EXEC stays all-1s, so the
// WMMA inside the taken path remains legal).
//
// Epilogue per element is 5 VALU: inline-asm v_med3_f32 (eps clamp, immune
// to instcombine re-expansion), raw v_sqrt_f32 (TRANS), v_sub, v_max, v_fmac.
// Deterministic two-pass reduction (no float atomics).
// ---------------------------------------------------------------------------

typedef float v2f __attribute__((ext_vector_type(2)));
typedef float v8f __attribute__((ext_vector_type(8)));

#define MC_GRID_H 128
#define MC_GRID_W 128
#define MC_N_POS  8192
#define MC_EPS    1e-12f

#define MC_ROW_TILES      (MC_N_POS / 16)                 // 512
#define MC_COL_TILES      ((MC_GRID_H * MC_GRID_W) / 16)  // 1024
#define MC_COL_GROUPS     16
#define MC_TILES_PER_WAVE (MC_COL_TILES / MC_COL_GROUPS)  // 64
#define MC_WAVES_PER_BLK  8
#define MC_TOTAL_WAVES    (MC_ROW_TILES * MC_COL_GROUPS)  // 8192
#define MC_BLOCKS         (MC_TOTAL_WAVES / MC_WAVES_PER_BLK) // 1024

__global__ __launch_bounds__(256)
void MazeCollision_wmma_kernel(const float2* __restrict__ pos,
                               const float*  __restrict__ maze,
                               const float*  __restrict__ radp,
                               float*        __restrict__ ws)
{
    const int lane      = threadIdx.x & 31;
    const int waveInBlk = threadIdx.x >> 5;
    const int waveId    = blockIdx.x * MC_WAVES_PER_BLK + waveInBlk;
    const int rowTile   = waveId >> 4;               // / MC_COL_GROUPS
    const int colGroup  = waveId & (MC_COL_GROUPS - 1);
    const int rowBase   = rowTile * 16;
    const int l16       = lane & 15;
    const int half      = lane >> 4;
    const float rad     = *radp;

    // ---- A operand: 16x4 f32 tile of positions (row-major in lanes).
    // ISA layout: lanes 0-15 -> M=lane, VGPR0=K0, VGPR1=K1;
    //             lanes 16-31 -> K2/K3 (our zero padding).
    // Load unconditionally + select so EXEC never diverges (WMMA needs EXEC all-1s).
    const float2 pa = pos[rowBase + l16];
    const float px = pa.x;     // valid in ALL lanes (halves duplicate rows)
    const float py = pa.y;
    v2f a;
    a.x = (lane < 16) ? px : 0.0f;   // K=0 : p.i
    a.y = (lane < 16) ? py : 0.0f;   // K=1 : p.j

    // ---- |p|^2 for the 8 D-matrix rows this lane-half owns
    // (D layout: VGPR g, lanes 0-15 -> M=g ; lanes 16-31 -> M=g+8).
    float p2r[8];
#pragma unroll
    for (int g = 0; g < 8; ++g) {
        const float2 p = pos[rowBase + g + 8 * half];
        p2r[g] = __builtin_fmaf(p.x, p.x, p.y * p.y);
    }

    // Hoisted constants for the inline-asm med3 clamp.
    const float epsv = MC_EPS;
    const float infv = __builtin_inff();

    float acc = 0.0f;

    for (int t = 0; t < MC_TILES_PER_WAVE; ++t) {
        const int colBase = (colGroup * MC_TILES_PER_WAVE + t) * 16;
        const int i_row = colBase >> 7;            // maze row of this strip
        const int j0    = colBase & (MC_GRID_W - 1);
        const float ci    = (float)i_row + 0.5f;
        const float jbase = (float)j0;

        // ---- Conservative cull: does ANY position touch this strip?
        // Pair (p, cell) can contribute only if |p.x - ci| <= rad AND
        // p.y within [first_center - rad, last_center + rad]. Exact zero
        // otherwise (relu), so skipping is bit-safe for the sum.
        const bool near = (__builtin_fabsf(px - ci) <= rad) &
                          (py >= jbase + 0.5f - rad) &
                          (py <= jbase + 15.5f + rad);
        if (__builtin_amdgcn_ballot_w32(near) == 0u) continue;  // uniform branch

        const int m  = colBase + l16;        // flat cell index for this lane's column
        const float cj = (float)(j0 + l16) + 0.5f;
        const float occ = maze[m];           // L2-resident load, only on near tiles
        const float c2  = __builtin_fmaf(ci, ci, cj * cj);

        // ---- B operand: 4x16 f32, fold the -2 in. lanes 0-15 = K0/K1 rows,
        // lanes 16-31 = K2/K3 zero-pad (mirrors the documented A layout).
        v2f b;
        b.x = (lane < 16) ? (-2.0f * ci) : 0.0f;   // K=0 row, N=lane
        b.y = (lane < 16) ? (-2.0f * cj) : 0.0f;   // K=1 row, N=lane

        // ---- C operand: |p|^2_row + |c|^2_col so D = d^2 directly.
        v8f c;
#pragma unroll
        for (int g = 0; g < 8; ++g) c[g] = p2r[g] + c2;

        // D(16x16,f32) = A(16x4) x B(4x16) + C
        v8f d2 = __builtin_amdgcn_wmma_f32_16x16x4_f32(
            /*neg_a=*/false, a, /*neg_b=*/false, b,
            /*c_mod=*/(short)0, c, /*reuse_a=*/false, /*reuse_b=*/false);

        // ---- epilogue: pen = relu(rad - sqrt(clamp(d2, eps, inf))) * occ
#pragma unroll
        for (int g = 0; g < 8; ++g) {
            float d2c;
            // clamp(d2, eps, +inf) == max(d2, eps) for finite d2, in exactly
            // one v_med3_f32 (inline asm so LLVM can't re-expand it).
            asm("v_med3_f32 %0, %1, %2, %3"
                : "=v"(d2c) : "v"(d2[g]), "v"(epsv), "v"(infv));
            const float d   = __builtin_amdgcn_sqrtf(d2c);   // single v_sqrt_f32
            const float pen = __builtin_fmaxf(rad - d, 0.0f);
            acc = __builtin_fmaf(pen, occ, acc);
        }
    }

    // ---- wave32 reduction (fixed order -> deterministic)
#pragma unroll
    for (int off = 16; off > 0; off >>= 1)
        acc += __shfl_xor(acc, off, 32);

    if (lane == 0) ws[waveId] = acc;
}

// Deterministic final reduction: one block, fixed traversal order.
__global__ __launch_bounds__(256)
void MazeCollision_reduce_kernel(const float* __restrict__ ws,
                                 float* __restrict__ out)
{
    __shared__ float s[256];
    float v = 0.0f;
    for (int i = threadIdx.x; i < MC_TOTAL_WAVES; i += 256) v += ws[i];
    s[threadIdx.x] = v;
    __syncthreads();
    for (int stride = 128; stride > 0; stride >>= 1) {
        if (threadIdx.x < stride) s[threadIdx.x] += s[threadIdx.x + stride];
        __syncthreads();
    }
    if (threadIdx.x == 0) out[0] = s[0];
}

extern "C" void kernel_launch(void* const* d_in, const int* in_sizes, int n_in,
                              void* d_out, int out_size, void* d_ws, size_t ws_size,
                              hipStream_t stream)
{
    (void)in_sizes; (void)n_in; (void)out_size; (void)ws_size;
    const float2* pos  = (const float2*)d_in[0];   // [N, 2] f32
    const float*  maze = (const float*)d_in[1];    // [H, W] f32
    const float*  radp = (const float*)d_in[2];    // scalar f32
    float*        out  = (float*)d_out;            // [1] f32
    float*        ws   = (float*)d_ws;             // MC_TOTAL_WAVES floats (32 KB)

    MazeCollision_wmma_kernel<<<MC_BLOCKS, 256, 0, stream>>>(pos, maze, radp, ws);
    MazeCollision_reduce_kernel<<<1, 256, 0, stream>>>(ws, out);
}